// Arch_17918603559425
// MI455X (gfx1250) — compile-verified
//
#include <hip/hip_runtime.h>
#include <hip/hip_bf16.h>

// ---------------------------------------------------------------------------
// Problem constants (from the reference)
// ---------------------------------------------------------------------------
#define TT   457          // SEQ
#define BB   4096         // batch
#define HH   5            // hidden == input
#define GG   20           // 4*H gates
#define NL   10           // layers
#define KK   2285         // T*H  (FC reduction dim)
#define OUTN 457          // FC output dim

typedef __attribute__((ext_vector_type(2))) float v2f;
typedef __attribute__((ext_vector_type(8))) float v8f;

__device__ __forceinline__ float fast_sigmoid(float x) {
    return 1.0f / (1.0f + __expf(-x));
}
__device__ __forceinline__ float fast_tanh(float x) {
    // 1 - 2/(e^{2x}+1): monotone, saturates correctly at +/-1 for |x| large
    float e = __expf(2.0f * x);
    return 1.0f - 2.0f / (e + 1.0f);
}

// ---------------------------------------------------------------------------
// LSTM layer: 8 lanes per sample (5 active hidden units, lanes 5..7 redundant
// but convergent). Per-lane state h[k], c[k]; cross-lane broadcast with
// ds_bpermute (no LDS memory, no barriers).
//
// Software pipelining: the x-projection for step t+1 (load + broadcast +
// 5 FMAs per gate) is h-independent and computed off the critical path;
// the recurrent chain per step is only: bpermute(h) -> 5 FMAs -> activations.
// ---------------------------------------------------------------------------
__global__ __launch_bounds__(256)
void lstm_layer_kernel(const float* __restrict__ in, float* __restrict__ out,
                       const float* __restrict__ Wih, const float* __restrict__ Whh,
                       const float* __restrict__ bih, const float* __restrict__ bhh,
                       int in_ts, int in_bs, int out_ts, int out_bs)
{
    const int g      = blockIdx.x * blockDim.x + threadIdx.x;
    const int sample = g >> 3;
    const int sub    = g & 7;
    const int keff   = (sub < HH) ? sub : (sub - HH);     // lanes 5..7 duplicate 0..2
    const int lbase  = (int)(threadIdx.x & 24);           // subgroup base lane in wave

    // 4 gate rows (i,f,g,o for unit keff): Wih part, Whh part, fused bias
    float wx[4][HH], wh[4][HH], bz[4];
#pragma unroll
    for (int q = 0; q < 4; ++q) {
        const int j = keff + HH * q;
#pragma unroll
        for (int i = 0; i < HH; ++i) wx[q][i] = Wih[j * HH + i];
#pragma unroll
        for (int i = 0; i < HH; ++i) wh[q][i] = Whh[j * HH + i];
        bz[q] = bih[j] + bhh[j];
    }

    const float* __restrict__ inp = in + (size_t)sample * in_bs + keff;
    float*       __restrict__ op  = out + (size_t)sample * out_bs + sub;

    float h = 0.0f, c = 0.0f;

    // ---- prologue: x-projection for t = 0 (all lanes load; 5..7 duplicate) --
    float xpart[4];
    {
        const int xb = __float_as_int(inp[0]);
        float zx[HH];
#pragma unroll
        for (int i = 0; i < HH; ++i)
            zx[i] = __int_as_float(__builtin_amdgcn_ds_bpermute((lbase + i) << 2, xb));
#pragma unroll
        for (int q = 0; q < 4; ++q) {
            float a = bz[q];
#pragma unroll
            for (int i = 0; i < HH; ++i) a = fmaf(wx[q][i], zx[i], a);
            xpart[q] = a;
        }
    }

    for (int t = 0; t < TT; ++t) {
        // prefetch x for t+1 (clamped; value unused on last step)
        const int tn = (t + 1 < TT) ? (t + 1) : (TT - 1);
        const float xn = inp[(size_t)tn * in_ts];

        // ---- recurrent critical path: broadcast h, 5 FMAs, activations ----
        const int hb = __float_as_int(h);
        float zh[HH];
#pragma unroll
        for (int i = 0; i < HH; ++i)
            zh[i] = __int_as_float(__builtin_amdgcn_ds_bpermute((lbase + i) << 2, hb));

        float gate[4];
#pragma unroll
        for (int q = 0; q < 4; ++q) {
            float a = xpart[q];
#pragma unroll
            for (int i = 0; i < HH; ++i) a = fmaf(wh[q][i], zh[i], a);
            gate[q] = a;
        }

        const float ig = fast_sigmoid(gate[0]);
        const float fg = fast_sigmoid(gate[1]);
        const float gg = fast_tanh(gate[2]);
        const float og = fast_sigmoid(gate[3]);
        c = fmaf(fg, c, ig * gg);
        h = og * fast_tanh(c);

        if (sub < HH)
            op[(size_t)t * out_ts] = h;

        // ---- off-critical-path: x-projection for t+1 ----
        const int xb = __float_as_int(xn);
        float zx[HH];
#pragma unroll
        for (int i = 0; i < HH; ++i)
            zx[i] = __int_as_float(__builtin_amdgcn_ds_bpermute((lbase + i) << 2, xb));
#pragma unroll
        for (int q = 0; q < 4; ++q) {
            float a = bz[q];
#pragma unroll
            for (int i = 0; i < HH; ++i) a = fmaf(wx[q][i], zx[i], a);
            xpart[q] = a;
        }
    }
}

// ---------------------------------------------------------------------------
// FC: out[4096,457] = A[4096,2285] * fc_W[457,2285]^T + fc_b
// One wave per 32x16 output tile (two 16x16 M-tiles sharing the B operand),
// V_WMMA_F32_16X16X4_F32 over K in chunks of 4.
//   A layout : lane l holds A[m0+(l&15)][k0 + 2*(l>>4) + {0,1}] in 2 VGPRs
//   B layout : lane l holds B[k0 + 2*(l>>4) + {0,1}][n0+(l&15)] (= fc_W[n][k])
//   D layout : VGPR r, lane l -> row m0 + r + 8*(l>>4), col n0 + (l&15)
// ---------------------------------------------------------------------------
__global__ __launch_bounds__(32)
void fc_wmma_kernel(const float* __restrict__ A, const float* __restrict__ W,
                    const float* __restrict__ bias, float* __restrict__ out)
{
    const int m0   = blockIdx.x << 5;            // two 16-row tiles: m0, m0+16
    const int n0   = blockIdx.y << 4;
    const int lane = (int)(threadIdx.x & 31);
    const int hi   = lane >> 4;
    const int lid  = lane & 15;
    const int n    = n0 + lid;
    const bool nv  = (n < OUTN);

    const float* __restrict__ Ar0 = A + (size_t)(m0 + lid) * KK;
    const float* __restrict__ Ar1 = Ar0 + (size_t)16 * KK;
    const float* __restrict__ Wr  = W + (size_t)(nv ? n : 0) * KK;
    const int koff = hi << 1;

    v8f acc0 = {};
    v8f acc1 = {};

    // 571 full K-chunks of 4 (K = 2284 covered)
    for (int k0 = 0; k0 < KK - 1; k0 += 4) {
        v2f a0, a1, b;
        a0[0] = Ar0[k0 + koff];
        a0[1] = Ar0[k0 + koff + 1];
        a1[0] = Ar1[k0 + koff];
        a1[1] = Ar1[k0 + koff + 1];
        const float b0 = Wr[k0 + koff];
        const float b1 = Wr[k0 + koff + 1];
        b[0] = nv ? b0 : 0.0f;
        b[1] = nv ? b1 : 0.0f;
        acc0 = __builtin_amdgcn_wmma_f32_16x16x4_f32(false, a0, false, b,
                                                     (short)0, acc0, false, false);
        acc1 = __builtin_amdgcn_wmma_f32_16x16x4_f32(false, a1, false, b,
                                                     (short)0, acc1, false, false);
    }
    // tail: k = 2284 only (zero-pad k = 2285..2287); EXEC is full at the WMMAs
    {
        v2f a0 = {0.0f, 0.0f};
        v2f a1 = {0.0f, 0.0f};
        v2f b  = {0.0f, 0.0f};
        if (hi == 0) {
            a0[0] = Ar0[KK - 1];
            a1[0] = Ar1[KK - 1];
            b[0]  = nv ? Wr[KK - 1] : 0.0f;
        }
        acc0 = __builtin_amdgcn_wmma_f32_16x16x4_f32(false, a0, false, b,
                                                     (short)0, acc0, false, false);
        acc1 = __builtin_amdgcn_wmma_f32_16x16x4_f32(false, a1, false, b,
                                                     (short)0, acc1, false, false);
    }

    if (nv) {
        const float bv = bias[n];
#pragma unroll
        for (int r = 0; r < 8; ++r) {
            const int mr = m0 + r + (hi << 3);
            out[(size_t)mr * OUTN + n]        = acc0[r] + bv;
            out[(size_t)(mr + 16) * OUTN + n] = acc1[r] + bv;
        }
    }
}

// ---------------------------------------------------------------------------
// Launch: 10 stream-ordered LSTM layer passes (ping-pong in d_ws), last layer
// stores directly in flat [B, T*H] layout, then one WMMA GEMM for the FC.
// d_ws usage: 2 * T*B*H floats = 74.9 MB.
// ---------------------------------------------------------------------------
extern "C" void kernel_launch(void* const* d_in, const int* in_sizes, int n_in,
                              void* d_out, int out_size, void* d_ws, size_t ws_size,
                              hipStream_t stream)
{
    (void)in_sizes; (void)n_in; (void)out_size; (void)ws_size;

    const float* x    = (const float*)d_in[0];   // [B, T, 5] f32
    const float* Wih  = (const float*)d_in[3];   // [10, 20, 5]
    const float* Whh  = (const float*)d_in[4];   // [10, 20, 5]
    const float* bih  = (const float*)d_in[5];   // [10, 20]
    const float* bhh  = (const float*)d_in[6];   // [10, 20]
    const float* fcW  = (const float*)d_in[7];   // [457, 2285]
    const float* fcb  = (const float*)d_in[8];   // [457]
    float*       out  = (float*)d_out;           // [4096, 457]

    float* buf[2];
    buf[0] = (float*)d_ws;
    buf[1] = buf[0] + (size_t)TT * BB * HH;

    const dim3 lblk(256);
    const dim3 lgrid((BB * 8) / 256);            // 8 lanes per sample -> 128 blocks

    const float* cur = x;
    int in_ts = HH, in_bs = TT * HH;             // x: b*2285 + t*5 + i
    for (int l = 0; l < NL; ++l) {
        float* o = buf[l & 1];
        int out_ts, out_bs;
        if (l == NL - 1) { out_ts = HH;      out_bs = TT * HH; } // flat [B, T*H]
        else             { out_ts = BB * HH; out_bs = HH;      } // [T][B][H]
        lstm_layer_kernel<<<lgrid, lblk, 0, stream>>>(
            cur, o,
            Wih + (size_t)l * GG * HH, Whh + (size_t)l * GG * HH,
            bih + (size_t)l * GG,      bhh + (size_t)l * GG,
            in_ts, in_bs, out_ts, out_bs);
        cur = o; in_ts = out_ts; in_bs = out_bs;
    }

    const dim3 fgrid(BB / 32, (OUTN + 15) / 16); // 128 x 29 tiles, 1 wave each
    fc_wmma_kernel<<<fgrid, dim3(32), 0, stream>>>(cur, fcW, fcb, out);
}